// BoundingBoxMatcher_75084618268928
// MI455X (gfx1250) — compile-verified
//
#include <hip/hip_runtime.h>
#include <math.h>

#define B_  8
#define N_  65536
#define H_  32

typedef float v2f __attribute__((ext_vector_type(2)));
typedef float v8f __attribute__((ext_vector_type(8)));

#define LOG2E_F  1.44269504088896340736f
#define LN2_F    0.69314718055994530942f

// ---- global workspace layout (floats) ----
#define WS_CROSS 0
#define WS_TP    (B_*H_*H_)
#define WS_SP    (2*B_*H_*H_)
#define WS_PROB  (WS_SP + B_*H_)
#define WS_INS   (WS_PROB + B_*H_)
#define WS_ACC_TOTAL (WS_INS + B_*H_)
#define WS_SCORE WS_ACC_TOTAL
#define WS_TOTAL (WS_SCORE + B_*H_*H_)

// ---- per-workgroup LDS accumulator layout (floats) ----
#define SA_CROSS 0
#define SA_TP    (H_*H_)
#define SA_SP    (2*H_*H_)
#define SA_PROB  (SA_SP + H_)
#define SA_INS   (SA_PROB + H_)
#define SA_TOTAL (SA_INS + H_)

__global__ void zero_acc_kernel(float* ws) {
  int i = blockIdx.x * blockDim.x + threadIdx.x;
  if (i < WS_ACC_TOTAL) ws[i] = 0.f;
}

__device__ __forceinline__ float insideVal(const float* bx, float x, float y, float z) {
  float p0 = (bx[0] - x) * (bx[3] - x);
  float p1 = (bx[1] - y) * (bx[4] - y);
  float p2 = (bx[2] - z) * (bx[5] - z);
  // all(p<=0)  <=>  max3(p0,p1,p2) <= 0
  return (fmaxf(fmaxf(p0, p1), p2) <= 0.f) ? 1.f : 0.f;
}

// logit, sigmoid(logit), softplus(logit) with raw v_exp_f32 / v_log_f32.
// l is clamped to [-20,20], so e = exp(-|l|) in [2e-9, 1] and 1+e in [1,2]:
// no overflow/underflow/denormal cases -> raw HW transcendentals are safe.
__device__ __forceinline__ void kVals(const float* bx, float x, float y, float z,
                                      float& lg, float& sg, float& sp) {
  float p0 = (bx[0] - x) * (bx[3] - x);
  float p1 = (bx[1] - y) * (bx[4] - y);
  float p2 = (bx[2] - z) * (bx[5] - z);
  float l0 = fminf(fmaxf(-100.f * p0, -20.f), 20.f);
  float l1 = fminf(fmaxf(-100.f * p1, -20.f), 20.f);
  float l2 = fminf(fmaxf(-100.f * p2, -20.f), 20.f);
  float l  = fminf(l0, fminf(l1, l2));
  float e  = __builtin_amdgcn_exp2f(-fabsf(l) * LOG2E_F);  // v_exp_f32
  float u  = 1.f + e;
  float r  = __builtin_amdgcn_rcpf(u);                     // v_rcp_f32
  lg = l;
  sg = (l >= 0.f) ? r : e * r;                             // sigmoid(l)
  sp = fmaxf(l, 0.f) + LN2_F * __builtin_amdgcn_logf(u);   // softplus(l), v_log_f32
}

// Big reduction over N: cross[h][k], tp[h][k], sums of softplus / prob / inside.
// The bhn,bkn->bhk einsums run on v_wmma_f32_16x16x4_f32 (f32-exact).
__global__ __launch_bounds__(256) void score_accum_kernel(
    const float* __restrict__ pc,   // (B, N, 9)
    const float* __restrict__ vt,   // (B, 32, 2, 3)
    const float* __restrict__ vp,   // (B, 32, 2, 3)
    float* __restrict__ ws) {
  const int b    = blockIdx.y;
  const int lane = threadIdx.x & 31;
  const int hh   = lane >> 4;     // 0: lanes 0-15, 1: lanes 16-31
  const int m    = lane & 15;

  __shared__ float sacc[SA_TOTAL];
  for (int i = threadIdx.x; i < SA_TOTAL; i += blockDim.x) sacc[i] = 0.f;
  __syncthreads();

  // Per-lane box params: true boxes h=m, m+16 (A tiles); pred boxes k=m, m+16 (B tiles)
  float vtLo[6], vtHi[6], vpLo[6], vpHi[6];
  const float* vtb = vt + b * H_ * 6;
  const float* vpb = vp + b * H_ * 6;
#pragma unroll
  for (int c = 0; c < 6; ++c) {
    vtLo[c] = vtb[m * 6 + c];        vtHi[c] = vtb[(m + 16) * 6 + c];
    vpLo[c] = vpb[m * 6 + c];        vpHi[c] = vpb[(m + 16) * 6 + c];
  }

  v8f cr00 = {}, cr01 = {}, cr10 = {}, cr11 = {};  // cross = inside x logits
  v8f tq00 = {}, tq01 = {}, tq10 = {}, tq11 = {};  // tp    = inside x sigmoid
  float spLo = 0.f, spHi = 0.f;
  float prLo = 0.f, prHi = 0.f;
  float insLo = 0.f, insHi = 0.f;

  const float* pcb = pc + (size_t)b * N_ * 9;
  const int waveId = (int)(blockIdx.x * (blockDim.x >> 5) + (threadIdx.x >> 5));
  const int nWaves = (int)(gridDim.x * (blockDim.x >> 5));
  const int nTiles = N_ / 32;

  for (int t = waveId; t < nTiles; t += nWaves) {
    const float* pp = pcb + (size_t)(t * 32 + lane) * 9;
    float px = pp[0], py = pp[1], pz = pp[2];
    if (t + nWaves < nTiles)  // gfx1250 global_prefetch_b8
      __builtin_prefetch(pcb + (size_t)((t + nWaves) * 32 + lane) * 9, 0, 0);

#pragma unroll 2              // lean body: 2x for WMMA/VALU co-execution ILP
    for (int s = 0; s < 8; ++s) {   // 4 points per WMMA step (K=4)
      const int i0 = s * 4 + 2 * hh;   // A/B K layout: lanes 0-15 -> K{0,1}, 16-31 -> K{2,3}
      float x0 = __shfl(px, i0, 32),     y0 = __shfl(py, i0, 32),     z0 = __shfl(pz, i0, 32);
      float x1 = __shfl(px, i0 + 1, 32), y1 = __shfl(py, i0 + 1, 32), z1 = __shfl(pz, i0 + 1, 32);

      v2f aL, aH, bL, bH, qL, qH;
      aL[0] = insideVal(vtLo, x0, y0, z0); aL[1] = insideVal(vtLo, x1, y1, z1);
      aH[0] = insideVal(vtHi, x0, y0, z0); aH[1] = insideVal(vtHi, x1, y1, z1);

      float lg, sg, sp;
      kVals(vpLo, x0, y0, z0, lg, sg, sp); bL[0] = lg; qL[0] = sg; spLo += sp;
      kVals(vpLo, x1, y1, z1, lg, sg, sp); bL[1] = lg; qL[1] = sg; spLo += sp;
      kVals(vpHi, x0, y0, z0, lg, sg, sp); bH[0] = lg; qH[0] = sg; spHi += sp;
      kVals(vpHi, x1, y1, z1, lg, sg, sp); bH[1] = lg; qH[1] = sg; spHi += sp;

      prLo += qL[0] + qL[1];  prHi += qH[0] + qH[1];
      insLo += aL[0] + aL[1]; insHi += aH[0] + aH[1];

      // D[h,k] += sum_K inside[h,K] * val[K,k]
      cr00 = __builtin_amdgcn_wmma_f32_16x16x4_f32(false, aL, false, bL, (short)0, cr00, false, false);
      cr01 = __builtin_amdgcn_wmma_f32_16x16x4_f32(false, aL, false, bH, (short)0, cr01, false, false);
      cr10 = __builtin_amdgcn_wmma_f32_16x16x4_f32(false, aH, false, bL, (short)0, cr10, false, false);
      cr11 = __builtin_amdgcn_wmma_f32_16x16x4_f32(false, aH, false, bH, (short)0, cr11, false, false);
      tq00 = __builtin_amdgcn_wmma_f32_16x16x4_f32(false, aL, false, qL, (short)0, tq00, false, false);
      tq01 = __builtin_amdgcn_wmma_f32_16x16x4_f32(false, aL, false, qH, (short)0, tq01, false, false);
      tq10 = __builtin_amdgcn_wmma_f32_16x16x4_f32(false, aH, false, qL, (short)0, tq10, false, false);
      tq11 = __builtin_amdgcn_wmma_f32_16x16x4_f32(false, aH, false, qH, (short)0, tq11, false, false);
    }
  }

  // ---- level 1: reduce into LDS (ds_add_f32 atomics) ----
  atomicAdd(&sacc[SA_SP   + m],      spLo);  atomicAdd(&sacc[SA_SP   + m + 16], spHi);
  atomicAdd(&sacc[SA_PROB + m],      prLo);  atomicAdd(&sacc[SA_PROB + m + 16], prHi);
  atomicAdd(&sacc[SA_INS  + m],      insLo); atomicAdd(&sacc[SA_INS  + m + 16], insHi);

#pragma unroll
  for (int r = 0; r < 8; ++r) {
    const int mr = r + 8 * hh;   // C/D layout: VGPR r, lanes 16-31 carry M+8
    const int nc = m;
    atomicAdd(&sacc[SA_CROSS + (mr)      * H_ + nc],      cr00[r]);
    atomicAdd(&sacc[SA_CROSS + (mr)      * H_ + nc + 16], cr01[r]);
    atomicAdd(&sacc[SA_CROSS + (mr + 16) * H_ + nc],      cr10[r]);
    atomicAdd(&sacc[SA_CROSS + (mr + 16) * H_ + nc + 16], cr11[r]);
    atomicAdd(&sacc[SA_TP    + (mr)      * H_ + nc],      tq00[r]);
    atomicAdd(&sacc[SA_TP    + (mr)      * H_ + nc + 16], tq01[r]);
    atomicAdd(&sacc[SA_TP    + (mr + 16) * H_ + nc],      tq10[r]);
    atomicAdd(&sacc[SA_TP    + (mr + 16) * H_ + nc + 16], tq11[r]);
  }
  __syncthreads();

  // ---- level 2: one global-atomic flush per workgroup ----
  for (int i = threadIdx.x; i < SA_TOTAL; i += blockDim.x) {
    float v = sacc[i];
    if (i < SA_TP)
      atomicAdd(&ws[WS_CROSS + b * H_ * H_ + i], v);
    else if (i < SA_SP)
      atomicAdd(&ws[WS_TP + b * H_ * H_ + (i - SA_TP)], v);
    else if (i < SA_PROB)
      atomicAdd(&ws[WS_SP + b * H_ + (i - SA_SP)], v);
    else if (i < SA_INS)
      atomicAdd(&ws[WS_PROB + b * H_ + (i - SA_PROB)], v);
    else
      atomicAdd(&ws[WS_INS + b * H_ + (i - SA_INS)], v);
  }
}

// Finalize: score = bce + iou + mse, global max, mask, then Jonker-Volgenant
// (f64 replica of the reference) per batch.
__global__ __launch_bounds__(256) void finalize_kernel(
    const float* __restrict__ vt, const float* __restrict__ vp,
    float* __restrict__ ws, int* __restrict__ out) {
  const int tid = threadIdx.x;
  __shared__ float smax[256];
  const float invN = 1.0f / (float)N_;

  float lmax = -INFINITY;
  for (int idx = tid; idx < B_ * H_ * H_; idx += 256) {
    const int b = idx >> 10, h = (idx >> 5) & 31, k = idx & 31;
    float crossv = ws[WS_CROSS + idx] * invN;
    float spv    = ws[WS_SP + b * H_ + k] * invN;
    float bce    = spv - crossv;
    float tpv    = ws[WS_TP + idx];
    float denom  = ws[WS_PROB + b * H_ + k] + ws[WS_INS + b * H_ + h] - tpv;
    float iou    = (denom == 0.f) ? 0.f : -(tpv / denom);
    float mse = 0.f;
    const float* vth = vt + (b * H_ + h) * 6;
    const float* vpk = vp + (b * H_ + k) * 6;
#pragma unroll
    for (int c = 0; c < 6; ++c) { float d = vth[c] - vpk[c]; mse += d * d; }
    mse *= (1.f / 6.f);
    float sc = bce + iou + mse;
    ws[WS_SCORE + idx] = sc;
    lmax = fmaxf(lmax, sc);
  }
  smax[tid] = lmax;
  __syncthreads();
  for (int s = 128; s > 0; s >>= 1) {
    if (tid < s) smax[tid] = fmaxf(smax[tid], smax[tid + s]);
    __syncthreads();
  }
  const float maxLoss = 2.f * smax[0];

  // mask: rows with all-zero true vertices get the constant max_loss
  for (int idx = tid; idx < B_ * H_; idx += 256) {
    const int b = idx >> 5, h = idx & 31;
    float s = 0.f;
    const float* vth = vt + (b * H_ + h) * 6;
#pragma unroll
    for (int c = 0; c < 6; ++c) s += fabsf(vth[c]);
    if (s == 0.f)
      for (int k = 0; k < H_; ++k) ws[WS_SCORE + (b * H_ + h) * H_ + k] = maxLoss;
  }
  __syncthreads();

  // serial JV shortest-augmenting-path per batch, f64 like the reference
  if (tid < B_) {
    const int b = tid;
    const float* cost = ws + WS_SCORE + b * H_ * H_;
    const double DINF = __builtin_inf();
    double u[H_ + 1], v[H_ + 1], minv[H_ + 1];
    int p[H_ + 1], way[H_ + 1];
    bool used[H_ + 1];
    for (int j = 0; j <= H_; ++j) { u[j] = 0.0; v[j] = 0.0; p[j] = 0; way[j] = 0; }

    for (int i = 1; i <= H_; ++i) {
      p[0] = i;
      int j0 = 0;
      for (int j = 0; j <= H_; ++j) { minv[j] = DINF; used[j] = false; }
      while (true) {
        used[j0] = true;
        const int i0 = p[j0];
        double delta = DINF;
        int j1 = 1;
        for (int j = 1; j <= H_; ++j) {
          if (!used[j]) {
            double cur = (double)cost[(i0 - 1) * H_ + (j - 1)] - u[i0] - v[j];
            if (cur < minv[j]) { minv[j] = cur; way[j] = j0; }
            if (minv[j] < delta) { delta = minv[j]; j1 = j; }  // strict <: first-min tie-break
          }
        }
        for (int j = 0; j <= H_; ++j) {
          if (used[j]) { u[p[j]] += delta; v[j] -= delta; }
          else          minv[j] -= delta;
        }
        j0 = j1;
        if (p[j0] == 0) break;
      }
      while (j0) { const int jprev = way[j0]; p[j0] = p[jprev]; j0 = jprev; }
    }
    for (int j = 1; j <= H_; ++j) out[b * H_ + (p[j] - 1)] = j - 1;
  }
}

extern "C" void kernel_launch(void* const* d_in, const int* in_sizes, int n_in,
                              void* d_out, int out_size, void* d_ws, size_t ws_size,
                              hipStream_t stream) {
  (void)in_sizes; (void)n_in; (void)out_size; (void)ws_size;
  const float* pc = (const float*)d_in[0];   // (8, 65536, 9) f32
  const float* vt = (const float*)d_in[1];   // (8, 32, 2, 3) f32
  const float* vp = (const float*)d_in[2];   // (8, 32, 2, 3) f32
  float* ws = (float*)d_ws;
  int* out = (int*)d_out;                    // (8, 32) int32

  zero_acc_kernel<<<(WS_ACC_TOTAL + 255) / 256, 256, 0, stream>>>(ws);
  score_accum_kernel<<<dim3(64, B_, 1), 256, 0, stream>>>(pc, vt, vp, ws);
  finalize_kernel<<<1, 256, 0, stream>>>(vt, vp, ws, out);
}